// GNN_17781164605813
// MI455X (gfx1250) — compile-verified
//
#include <hip/hip_runtime.h>
#include <math.h>

typedef __attribute__((ext_vector_type(2))) float v2f;
typedef __attribute__((ext_vector_type(8))) float v8f;
typedef unsigned int u32x4 __attribute__((ext_vector_type(4)));
typedef int i32x8 __attribute__((ext_vector_type(8)));
typedef int i32x4 __attribute__((ext_vector_type(4)));

#define N_NODES 32768
#define N_EDGES 262144
#define N_TOT   (N_EDGES + N_NODES)   // edges + self loops
#define HID 256
#define HEADS 4
#define CH 64
#define BGRAPH 64
#define NODES_PER_G (N_NODES / BGRAPH)
#define AS_STRIDE 260                 // 256 + 4 pad DWORDs inserted by TDM

__device__ __forceinline__ float gelu_exact(float x) {
  return 0.5f * x * (1.0f + erff(x * 0.70710678118654752f));
}
__device__ __forceinline__ float lrelu(float x) { return x > 0.f ? x : 0.2f * x; }

// monotonic float<->uint encoding so unsigned atomicMax == float max
__device__ __forceinline__ unsigned int enc_ord(float f) {
  unsigned int u = __float_as_uint(f);
  return (u & 0x80000000u) ? ~u : (u | 0x80000000u);
}
__device__ __forceinline__ float dec_ord(unsigned int e) {
  unsigned int u = (e & 0x80000000u) ? (e & 0x7fffffffu) : ~e;
  return __uint_as_float(u);
}

// ---------------------------------------------------------------------------
// Zero accumulator, init segment-max (encoded -inf) and denom
__global__ void k_init(float* __restrict__ acc, unsigned int* __restrict__ smax,
                       float* __restrict__ denom) {
  int i = blockIdx.x * 256 + threadIdx.x;
  if (i < N_NODES * HID) acc[i] = 0.f;
  if (i < N_NODES * HEADS) { smax[i] = 0x007fffffu; /* enc(-inf) */ denom[i] = 0.f; }
}

// ---------------------------------------------------------------------------
// Dual GEMM: Y0 = X @ W0, Y1 = X @ W1   (X:[N,256], W:[256,256], fp32 WMMA)
// Block = 256 thr (8 waves) handles a 16-row strip of X across all 256 cols.
// A tile staged via the Tensor Data Mover (TDM): one descriptor DMA moves the
// 16x256 fp32 tile global->LDS with 4-DWORD padding per 256-DWORD row, giving
// a 260-float row stride (bank step 4 -> conflict-free ds_load_b64 fragments).
__global__ void __launch_bounds__(256) k_gemm_dual(
    const float* __restrict__ X, const float* __restrict__ W0,
    const float* __restrict__ W1, float* __restrict__ Y0, float* __restrict__ Y1) {
  __shared__ float As[16 * AS_STRIDE];           // [m][k padded], ~16.6 KB
  const int mbase = blockIdx.x * 16;
  const int t = threadIdx.x;

  if (t < 32) {                                  // wave 0 issues the TDM DMA
    unsigned long long ga = (unsigned long long)(const void*)(X + (size_t)mbase * HID);
    unsigned lds = (unsigned)(size_t)(void*)As;  // LDS aperture: low 32 bits = LDS offset
    u32x4 g0;
    g0[0] = 1u;                                  // count=1 (valid), load, no gather
    g0[1] = lds;                                 // lds_addr
    g0[2] = (unsigned)(ga & 0xffffffffu);        // global_addr[31:0]
    g0[3] = (unsigned)((ga >> 32) & 0x01ffffffu) // global_addr[56:32]
            | 0x80000000u;                       // type=2 ("image")
    i32x8 g1;
    g1[0] = 0x07D20000;       // data_size=4B | pad_enable | pad_interval=256dw | pad_amount=4dw
    g1[1] = (int)(256u << 16);   // tensor_dim0 = 256 (bits 79:48, low half)
    g1[2] = (int)(32768u << 16); // tensor_dim1 = 32768 (bits 111:80, low half)
    g1[3] = (int)(256u << 16);   // tile_dim0 = 256 (bits 127:112)
    g1[4] = 16;                  // tile_dim1 = 16 (bits 143:128); tile_dim2 = 0
    g1[5] = 256;                 // tensor_dim0_stride = 256 (bits 207:160, low 32)
    g1[6] = 0;                   // stride hi / tensor_dim1_stride lo (unused, 2D)
    g1[7] = 0;
    i32x4 z4 = {0, 0, 0, 0};                     // groups 2/3 unused (<=2D tensor)
    i32x8 z8 = {0, 0, 0, 0, 0, 0, 0, 0};         // extra group (6-arg toolchain form)
    __builtin_amdgcn_tensor_load_to_lds(g0, g1, z4, z4, z8, 0);
    __builtin_amdgcn_s_wait_tensorcnt(0);        // TENSORcnt -> 0 before publishing
  }
  __syncthreads();

  const int lane = t & 31;
  const int wid  = t >> 5;
  const int mi   = lane & 15;
  const int koff = (lane >> 4) * 2;              // A frag: VGPR0=K0/K2, VGPR1=K1/K3
  const int n0   = wid * 16 + mi;                // wave owns ntiles wid and wid+8
  const int n1   = (wid + 8) * 16 + mi;

  v8f c00 = {}, c01 = {}, c10 = {}, c11 = {};
  for (int kk = 0; kk < HID; kk += 4) {
    const int k0 = kk + koff;
    v2f a = *(const v2f*)&As[mi * AS_STRIDE + k0];   // ds_load_b64, conflict-free
    v2f b;
    b.x = W0[k0 * HID + n0]; b.y = W0[(k0 + 1) * HID + n0];
    c00 = __builtin_amdgcn_wmma_f32_16x16x4_f32(false, a, false, b, (short)0, c00, false, false);
    b.x = W0[k0 * HID + n1]; b.y = W0[(k0 + 1) * HID + n1];
    c01 = __builtin_amdgcn_wmma_f32_16x16x4_f32(false, a, false, b, (short)0, c01, false, false);
    b.x = W1[k0 * HID + n0]; b.y = W1[(k0 + 1) * HID + n0];
    c10 = __builtin_amdgcn_wmma_f32_16x16x4_f32(false, a, false, b, (short)0, c10, false, false);
    b.x = W1[k0 * HID + n1]; b.y = W1[(k0 + 1) * HID + n1];
    c11 = __builtin_amdgcn_wmma_f32_16x16x4_f32(false, a, false, b, (short)0, c11, false, false);
  }

  const int rbase = mbase + (lane >> 4) * 8;     // C/D: VGPR i -> M=i (+8 for hi lanes)
#pragma unroll
  for (int i = 0; i < 8; ++i) {
    Y0[(rbase + i) * HID + n0] = c00[i];
    Y0[(rbase + i) * HID + n1] = c01[i];
    Y1[(rbase + i) * HID + n0] = c10[i];
    Y1[(rbase + i) * HID + n1] = c11[i];
  }
}

// ---------------------------------------------------------------------------
// Per-(edge,head) attention logit + segment max (gathers hit L2: xl/xr = 64MB)
__global__ void k_edge_score(const float* __restrict__ xl, const float* __restrict__ xr,
                             const float* __restrict__ att,
                             const int* __restrict__ src, const int* __restrict__ dst,
                             float* __restrict__ score, unsigned int* __restrict__ smax) {
  int idx = blockIdx.x * blockDim.x + threadIdx.x;
  if (idx >= N_TOT * HEADS) return;
  int i = idx >> 2, h = idx & 3;
  int s, d;
  if (i < N_EDGES) { s = src[i]; d = dst[i]; } else { s = d = i - N_EDGES; }
  const float4* pl = (const float4*)(xl + (size_t)s * HID + h * CH);
  const float4* pr = (const float4*)(xr + (size_t)d * HID + h * CH);
  const float4* pa = (const float4*)(att + h * CH);
  float sc = 0.f;
#pragma unroll
  for (int c = 0; c < CH / 4; ++c) {
    float4 l = pl[c], r = pr[c], a = pa[c];
    sc += a.x * lrelu(l.x + r.x) + a.y * lrelu(l.y + r.y)
        + a.z * lrelu(l.z + r.z) + a.w * lrelu(l.w + r.w);
  }
  score[idx] = sc;
  atomicMax(&smax[d * HEADS + h], enc_ord(sc));
}

// exp(score - max) and segment sum
__global__ void k_edge_exp(const int* __restrict__ dst,
                           float* __restrict__ score,
                           const unsigned int* __restrict__ smax,
                           float* __restrict__ denom) {
  int idx = blockIdx.x * blockDim.x + threadIdx.x;
  if (idx >= N_TOT * HEADS) return;
  int i = idx >> 2, h = idx & 3;
  int d = (i < N_EDGES) ? dst[i] : i - N_EDGES;
  float ex = expf(score[idx] - dec_ord(smax[d * HEADS + h]));
  score[idx] = ex;
  atomicAdd(&denom[d * HEADS + h], ex);
}

// wave-per-edge aggregation: lane-consecutive channels -> coalesced atomics
__global__ void k_edge_agg(const float* __restrict__ xl,
                           const int* __restrict__ src, const int* __restrict__ dst,
                           const float* __restrict__ score, const float* __restrict__ denom,
                           float* __restrict__ acc) {
  int gid = blockIdx.x * blockDim.x + threadIdx.x;
  int i = gid >> 5, lane = gid & 31;
  if (i >= N_TOT) return;
  int s, d;
  if (i < N_EDGES) { s = src[i]; d = dst[i]; } else { s = d = i - N_EDGES; }
  float al[HEADS];
#pragma unroll
  for (int h = 0; h < HEADS; ++h)
    al[h] = score[i * HEADS + h] / denom[d * HEADS + h];
  const float* pl = xl + (size_t)s * HID;
  float* pd = acc + (size_t)d * HID;
#pragma unroll
  for (int j = 0; j < 8; ++j) {
    int c = lane + j * 32;                 // lanes cover 128B-contiguous runs
    atomicAdd(&pd[c], al[c >> 6] * pl[c]);
  }
}

// bias + exact GELU, in place
__global__ void k_bias_act(float* __restrict__ acc, const float* __restrict__ bias) {
  int i = blockIdx.x * 256 + threadIdx.x;
  if (i >= N_NODES * HID) return;
  acc[i] = gelu_exact(acc[i] + bias[i & (HID - 1)]);
}

// per-graph mean pool (512 contiguous nodes per graph)
__global__ void k_pool(const float* __restrict__ h, float* __restrict__ mean_out) {
  int b = blockIdx.x, f = threadIdx.x;
  const float* p = h + (size_t)b * NODES_PER_G * HID + f;
  float sum = 0.f;
  for (int r = 0; r < NODES_PER_G; ++r) sum += p[(size_t)r * HID];
  mean_out[b * HID + f] = sum * (1.0f / NODES_PER_G);
}

// small dense layer on the 64 root rows: out[b,j] = act(in[b,:]@W + bias)
__global__ void k_dense(const float* __restrict__ in, int in_stride,
                        const float* __restrict__ W, const float* __restrict__ bias,
                        float* __restrict__ out, int K, int J, int act) {
  int idx = blockIdx.x * blockDim.x + threadIdx.x;
  if (idx >= BGRAPH * J) return;
  int b = idx / J, j = idx % J;
  const float* pi = in + (size_t)b * in_stride;
  float sum = bias[j];
  for (int k = 0; k < K; ++k) sum += pi[k] * W[(size_t)k * J + j];
  out[idx] = act ? gelu_exact(sum) : sum;
}

// ---------------------------------------------------------------------------
extern "C" void kernel_launch(void* const* d_in, const int* in_sizes, int n_in,
                              void* d_out, int out_size, void* d_ws, size_t ws_size,
                              hipStream_t stream) {
  const float* x   = (const float*)d_in[0];
  const int*   ei  = (const int*)d_in[1];
  const int*   src = ei;
  const int*   dst = ei + N_EDGES;
  const float* W1l = (const float*)d_in[4];
  const float* W1r = (const float*)d_in[5];
  const float* a1  = (const float*)d_in[6];
  const float* b1  = (const float*)d_in[7];
  const float* W2l = (const float*)d_in[8];
  const float* W2r = (const float*)d_in[9];
  const float* a2  = (const float*)d_in[10];
  const float* b2  = (const float*)d_in[11];
  const float* Wm1 = (const float*)d_in[12];
  const float* bm1 = (const float*)d_in[13];
  const float* Wm2 = (const float*)d_in[14];
  const float* bm2 = (const float*)d_in[15];
  const float* Wf  = (const float*)d_in[16];
  const float* bf  = (const float*)d_in[17];
  float* out = (float*)d_out;

  // workspace layout (floats): 3 big buffers reused across layers + small ones
  float* ws    = (float*)d_ws;
  float* bufA  = ws;                              // xl   (32 MB)
  float* bufB  = bufA + (size_t)N_NODES * HID;    // xr   (32 MB)
  float* bufC  = bufB + (size_t)N_NODES * HID;    // acc/h (32 MB)
  float* score = bufC + (size_t)N_NODES * HID;    // (E+N)*H
  unsigned int* smax = (unsigned int*)(score + (size_t)N_TOT * HEADS);
  float* denom = (float*)(smax + (size_t)N_NODES * HEADS);
  float* z1    = denom + (size_t)N_NODES * HEADS; // 64*1024
  float* z2    = z1 + BGRAPH * 4 * HID;           // 64*256

  dim3 blk(256);
  const int g_big = (N_NODES * HID + 255) / 256;  // 32768 blocks
  const int g_eh  = (N_TOT * HEADS + 255) / 256;  // 4608 blocks
  const int g_agg = (N_TOT * 32 + 255) / 256;     // 36864 blocks

  for (int layer = 0; layer < 2; ++layer) {
    const float* in  = layer ? bufC : x;          // gemm reads bufC BEFORE init zeroes it
    const float* Wl  = layer ? W2l : W1l;
    const float* Wr  = layer ? W2r : W1r;
    const float* att = layer ? a2 : a1;
    const float* bia = layer ? b2 : b1;
    k_gemm_dual<<<dim3(N_NODES / 16), blk, 0, stream>>>(in, Wl, Wr, bufA, bufB);
    k_init<<<g_big, blk, 0, stream>>>(bufC, smax, denom);
    k_edge_score<<<g_eh, blk, 0, stream>>>(bufA, bufB, att, src, dst, score, smax);
    k_edge_exp<<<g_eh, blk, 0, stream>>>(dst, score, smax, denom);
    k_edge_agg<<<g_agg, blk, 0, stream>>>(bufA, src, dst, score, denom, bufC);
    k_bias_act<<<g_big, blk, 0, stream>>>(bufC, bia);
  }

  // graph mean -> second half of output
  k_pool<<<BGRAPH, blk, 0, stream>>>(bufC, out + BGRAPH * HID);
  // root-node MLP -> first half of output (root = node 0 of each 512-node block)
  k_dense<<<(BGRAPH * 4 * HID + 255) / 256, blk, 0, stream>>>(
      bufC, NODES_PER_G * HID, Wm1, bm1, z1, HID, 4 * HID, 1);
  k_dense<<<(BGRAPH * HID + 255) / 256, blk, 0, stream>>>(
      z1, 4 * HID, Wm2, bm2, z2, 4 * HID, HID, 1);
  k_dense<<<(BGRAPH * HID + 255) / 256, blk, 0, stream>>>(
      z2, HID, Wf, bf, out, HID, HID, 0);
}